// RelGraphConvOps_10900626997971
// MI455X (gfx1250) — compile-verified
//
#include <hip/hip_runtime.h>

typedef float v2f __attribute__((ext_vector_type(2)));
typedef float v4f __attribute__((ext_vector_type(4)));
typedef float v8f __attribute__((ext_vector_type(8)));

#define N_NODES   50000
#define N_EDGES   800000
#define IN_FEAT   64
#define OUT_FEAT  64
#define NUM_RELS  8
#define NUM_BASES 2
#define KDIM      64    // GEMM reduction dim (in_feat)
#define CDIM      192   // combined output cols: 2*64 basis + 64 loop

// ---------------------------------------------------------------------------
// Kernel 1: T = feat @ [W_b0 | W_b1 | loop_weight]   (N x 192), fp32 WMMA.
//   cols [0,128)   -> Y (workspace), stored basis-interleaved: Y[n][o][b]
//   cols [128,192) -> d_out = feat@loop_weight + bias  (scatter target init)
// One wave computes one 16x16 tile with V_WMMA_F32_16X16X4_F32.
// Weights staged in LDS as K-pairs so each B fragment is one ds_load_b64.
// ---------------------------------------------------------------------------
__global__ __launch_bounds__(256) void rgcn_gemm_kernel(
    const float* __restrict__ feat, const float* __restrict__ W,
    const float* __restrict__ loopW, const float* __restrict__ bias,
    float* __restrict__ Y, float* __restrict__ out)
{
  // Wp[k2][c] = { Wbig[2*k2][c], Wbig[2*k2+1][c] }  -- 32*192*8 = 48 KB
  __shared__ v2f Wp[(KDIM / 2) * CDIM];
  const int tid = threadIdx.x;
  for (int i = tid; i < (KDIM / 2) * CDIM; i += 256) {
    const int k2 = i / CDIM, c = i % CDIM;
    v2f w;
    if (c < 128) {
      const int b = c >> 6, o = c & 63;
      const float* wb = W + b * (IN_FEAT * OUT_FEAT) + o;
      w.x = wb[(2 * k2) * OUT_FEAT];
      w.y = wb[(2 * k2 + 1) * OUT_FEAT];
    } else {
      const float* wl = loopW + (c - 128);
      w.x = wl[(2 * k2) * OUT_FEAT];
      w.y = wl[(2 * k2 + 1) * OUT_FEAT];
    }
    Wp[i] = w;
  }
  __syncthreads();

  const int wave  = tid >> 5;
  const int lane  = tid & 31;
  const int t     = blockIdx.x * 8 + wave;                 // tile id (wave-uniform)
  const int numTiles = (N_NODES / 16) * (CDIM / 16);       // 3125 * 12 = 37500
  if (t >= numTiles) return;                               // uniform per wave

  const int row0  = (t / (CDIM / 16)) * 16;
  const int c0    = (t % (CDIM / 16)) * 16;
  const int ln    = lane & 15;
  const int khalf = lane >> 4;          // 0: K pair {k,k+1}, 1: {k+2,k+3}
  const int mrow  = row0 + ln;          // A-fragment row for this lane

  v8f acc = {};
#pragma unroll
  for (int k = 0; k < KDIM; k += 4) {
    const int ka = k + 2 * khalf;
    // A: 16x4 f32, lane holds (M=ln, K=ka..ka+1) -> contiguous float2
    v2f a = *(const v2f*)(feat + (size_t)mrow * IN_FEAT + ka);
    // B: 4x16 f32; lane needs rows {ka, ka+1} at col c0+ln -> one packed float2
    v2f b = Wp[(k / 2 + khalf) * CDIM + c0 + ln];
    acc = __builtin_amdgcn_wmma_f32_16x16x4_f32(
        /*neg_a=*/false, a, /*neg_b=*/false, b,
        /*c_mod=*/(short)0, acc, /*reuse_a=*/false, /*reuse_b=*/false);
  }

  // C/D layout: VGPR j holds (M=row0+j, N=ln) for lanes 0-15, (M=row0+j+8) for 16-31
  if (c0 < 128) {
    // basis-interleaved store: column c = b*64+o  ->  row offset o*2 + b
    const int c = c0 + ln;
    const int off = ((c & 63) << 1) | (c >> 6);
#pragma unroll
    for (int j = 0; j < 8; ++j) {
      const int row = row0 + j + 8 * khalf;
      Y[(size_t)row * 128 + off] = acc[j];
    }
  } else {
    const float bb = bias[(c0 - 128) + ln];
#pragma unroll
    for (int j = 0; j < 8; ++j) {
      const int row = row0 + j + 8 * khalf;
      out[(size_t)row * OUT_FEAT + (c0 - 128) + ln] = acc[j] + bb;
    }
  }
}

// ---------------------------------------------------------------------------
// Kernel 2: edge scatter.  h[dst] += c0*Y[src,0,:] + c1*Y[src,1,:]
// One wave per edge; lane handles output features {2*lane, 2*lane+1}.
// Y is basis-interleaved, so each lane's 4 operands are one float4 load
// (wave reads one contiguous 512B line). 64 atomic f32 adds per edge.
// ---------------------------------------------------------------------------
__global__ __launch_bounds__(256) void rgcn_edge_kernel(
    const float* __restrict__ Y, const float* __restrict__ coeff,
    const int* __restrict__ src, const int* __restrict__ dst,
    const int* __restrict__ etypes, float* __restrict__ out)
{
  const int gid = blockIdx.x * 256 + threadIdx.x;
  const int e = gid >> 5;
  if (e >= N_EDGES) return;
  const int lane = gid & 31;

  const int s = src[e];
  const int d = dst[e];
  const int r = etypes[e];
  const float c0 = coeff[r * NUM_BASES + 0];
  const float c1 = coeff[r * NUM_BASES + 1];

  // q = { Y0[o], Y1[o], Y0[o+1], Y1[o+1] },  o = 2*lane
  const v4f q = *(const v4f*)(Y + (size_t)s * 128 + 4 * lane);
  const float vx = c0 * q.x + c1 * q.y;
  const float vy = c0 * q.z + c1 * q.w;

  float* p = out + (size_t)d * OUT_FEAT + 2 * lane;
  unsafeAtomicAdd(p, vx);
  unsafeAtomicAdd(p + 1, vy);
}

// ---------------------------------------------------------------------------
extern "C" void kernel_launch(void* const* d_in, const int* in_sizes, int n_in,
                              void* d_out, int out_size, void* d_ws, size_t ws_size,
                              hipStream_t stream) {
  (void)in_sizes; (void)n_in; (void)out_size; (void)ws_size;
  const float* feat   = (const float*)d_in[0];   // [N, 64]
  const float* W      = (const float*)d_in[1];   // [2, 64, 64]
  const float* coeff  = (const float*)d_in[2];   // [8, 2]
  const float* h_bias = (const float*)d_in[3];   // [64]
  const float* loopW  = (const float*)d_in[4];   // [64, 64]
  const int*   src    = (const int*)d_in[5];     // [E]
  const int*   dst    = (const int*)d_in[6];     // [E]
  const int*   etyp   = (const int*)d_in[7];     // [E]
  float*       out    = (float*)d_out;           // [N, 64]
  float*       Y      = (float*)d_ws;            // [N, 128] = 25.6 MB scratch

  // Phase 1: WMMA node transform; also initializes out = loop + bias.
  const int numTiles = (N_NODES / 16) * (CDIM / 16);      // 37500 tiles
  const int gemmBlocks = (numTiles + 7) / 8;              // 8 waves/block
  rgcn_gemm_kernel<<<gemmBlocks, 256, 0, stream>>>(feat, W, loopW, h_bias, Y, out);

  // Phase 2: per-edge scatter (1 wave/edge).
  const long long totalThreads = (long long)N_EDGES * 32;
  const int edgeBlocks = (int)((totalThreads + 255) / 256);
  rgcn_edge_kernel<<<edgeBlocks, 256, 0, stream>>>(Y, coeff, src, dst, etyp, out);
}